// SSKernelNPLR_28449863368712
// MI455X (gfx1250) — compile-verified
//
#include <hip/hip_runtime.h>
#include <hip/hip_bf16.h>
#include <cstdint>

// ---------------------------------------------------------------------------
// S4 SSKernelNPLR kernel generation for MI455X (gfx1250, wave32).
//   Stage 1: Cauchy sums + Woodbury -> scaled complex k_f, packed as a
//            row-major real matrix A (512 x 2052) in d_ws.
//   Stage 2: irfft as real GEMM A(512x2050) x DFTbasis(2050x2048) using
//            V_WMMA_F32_16X16X4_F32. A strip staged to LDS with
//            global_load_async_to_lds_b128 (ASYNCcnt); basis generated with
//            a complex-rotation recurrence re-seeded by exact sincos.
// ---------------------------------------------------------------------------

#define HH    256
#define NN    64
#define CHN   2
#define LKER  2048
#define LL2   1025          // L/2 + 1
#define KPAD  2052          // 2*LL2 rounded up to multiple of 4
#define NSTEP (KPAD / 4)    // 513 WMMA K-steps
#define CHUNK 64            // recurrence re-seed interval (steps)

typedef __attribute__((ext_vector_type(2))) float v2f;
typedef __attribute__((ext_vector_type(8))) float v8f;

struct c32 { float re, im; };
__device__ __forceinline__ c32 cmul(c32 a, c32 b) {
    return { a.re * b.re - a.im * b.im, a.re * b.im + a.im * b.re };
}
__device__ __forceinline__ c32 cadd(c32 a, c32 b) { return { a.re + b.re, a.im + b.im }; }
__device__ __forceinline__ c32 csub(c32 a, c32 b) { return { a.re - b.re, a.im - b.im }; }
__device__ __forceinline__ c32 cinv(c32 a) {
    float d = a.re * a.re + a.im * a.im;
    float r = 1.0f / d;
    return { a.re * r, -a.im * r };
}

// ---------------------------------------------------------------------------
// Kernel 1: per (h, l) Cauchy reduction + Woodbury.
// grid = (ceil(LL2/256), H), block = 256. Parameters staged through LDS.
// A[(c*H + h)][2l] = Re(kf_c), A[...][2l+1] = Im(kf_c); K pad columns zeroed.
// ---------------------------------------------------------------------------
__global__ void __launch_bounds__(256)
cauchy_woodbury_kernel(const float* __restrict__ w_re, const float* __restrict__ w_im,
                       const float* __restrict__ B_re, const float* __restrict__ B_im,
                       const float* __restrict__ P_re, const float* __restrict__ P_im,
                       const float* __restrict__ Q_re, const float* __restrict__ Q_im,
                       const float* __restrict__ C_re, const float* __restrict__ C_im,
                       const float* __restrict__ log_dt,
                       float* __restrict__ A)
{
    __shared__ float s[12 * NN];   // wre,wim,Bre,Bim,Pre,Pim,Qre,Qim,C0re,C0im,C1re,C1im

    const int h   = blockIdx.y;
    const int tid = threadIdx.x;

    for (int i = tid; i < 12 * NN; i += blockDim.x) {
        const int which = i / NN;
        const int n     = i % NN;
        const float* src;
        switch (which) {
            case 0:  src = w_re + h * NN;            break;
            case 1:  src = w_im + h * NN;            break;
            case 2:  src = B_re + h * NN;            break;
            case 3:  src = B_im + h * NN;            break;
            case 4:  src = P_re + h * NN;            break;
            case 5:  src = P_im + h * NN;            break;
            case 6:  src = Q_re + h * NN;            break;
            case 7:  src = Q_im + h * NN;            break;
            case 8:  src = C_re + h * NN;            break;   // C[0,h,:]
            case 9:  src = C_im + h * NN;            break;
            case 10: src = C_re + (HH + h) * NN;     break;   // C[1,h,:]
            default: src = C_im + (HH + h) * NN;     break;
        }
        s[which * NN + n] = src[n];
    }
    __syncthreads();

    const int l = blockIdx.x * blockDim.x + tid;
    if (l >= LL2) return;

    const float dt = __expf(log_dt[h]);

    // omega = exp(-2*pi*i*l/L);  z = 2(1-omega)/(1+omega)
    const float theta = (6.28318530717958647692f / (float)LKER) * (float)l;
    float sn, cs;
    __sincosf(theta, &sn, &cs);
    const c32 om   = { cs, -sn };
    const c32 den1 = { 1.0f + om.re, om.im };       // 1 + omega
    const c32 iden = cinv(den1);
    const c32 z    = cmul((c32){ 2.0f - 2.0f * om.re, -2.0f * om.im }, iden);

    c32 r00 = {0,0}, r01 = {0,0}, r02 = {0,0};
    c32 r10 = {0,0}, r11 = {0,0}, r12 = {0,0};

    #pragma unroll 4
    for (int n = 0; n < NN; ++n) {
        const c32 wd  = { dt * s[0 * NN + n], dt * s[1 * NN + n] };
        const c32 dlt = { z.re - wd.re, z.im - wd.im };
        const c32 inv = cinv(dlt);

        const c32 Bv = { s[2 * NN + n],  s[3 * NN + n]  };
        const c32 Pv = { s[4 * NN + n],  s[5 * NN + n]  };
        const c32 Qv = { s[6 * NN + n],  s[7 * NN + n]  };
        const c32 C0 = { s[8 * NN + n],  s[9 * NN + n]  };
        const c32 C1 = { s[10 * NN + n], s[11 * NN + n] };

        const c32 t0 = cmul(Bv, inv);   // Bc[0] / (z - wd)
        const c32 t1 = cmul(Pv, inv);   // Bc[1] / (z - wd)

        r00 = cadd(r00, cmul(t0, C0));
        r01 = cadd(r01, cmul(t0, C1));
        r02 = cadd(r02, cmul(t0, Qv));
        r10 = cadd(r10, cmul(t1, C0));
        r11 = cadd(r11, cmul(t1, C1));
        r12 = cadd(r12, cmul(t1, Qv));
    }

    // r *= dt
    r00.re *= dt; r00.im *= dt;  r01.re *= dt; r01.im *= dt;  r02.re *= dt; r02.im *= dt;
    r10.re *= dt; r10.im *= dt;  r11.re *= dt; r11.im *= dt;  r12.re *= dt; r12.im *= dt;

    // Woodbury rank-1: kf_b = r0b - r02 * r1b / (1 + r12)
    const c32 fac = cmul(r02, cinv((c32){ 1.0f + r12.re, r12.im }));
    c32 kf0 = csub(r00, cmul(fac, r10));
    c32 kf1 = csub(r01, cmul(fac, r11));

    // * 2/(1+omega), folded with irfft hermitian weight (1/L ends, 2/L interior)
    const float wgt = (l == 0 || l == LKER / 2) ? (1.0f / (float)LKER) : (2.0f / (float)LKER);
    const c32 sc = { 2.0f * wgt * iden.re, 2.0f * wgt * iden.im };
    kf0 = cmul(kf0, sc);
    kf1 = cmul(kf1, sc);

    float* row0 = A + (size_t)(0 * HH + h) * KPAD;
    float* row1 = A + (size_t)(1 * HH + h) * KPAD;
    row0[2 * l]     = kf0.re;
    row0[2 * l + 1] = kf0.im;
    row1[2 * l]     = kf1.re;
    row1[2 * l + 1] = kf1.im;
    if (l == 0) {   // zero the K padding columns (2050, 2051)
        row0[2050] = 0.0f; row0[2051] = 0.0f;
        row1[2050] = 0.0f; row1[2051] = 0.0f;
    }
}

// ---------------------------------------------------------------------------
// Kernel 2: irfft as real DFT GEMM on the f32 WMMA pipe.
//   out[m][t] = sum_l  A[m][2l]*cos(2*pi*l*t/L) - A[m][2l+1]*sin(2*pi*l*t/L)
// Block: one 16-row A strip (staged to LDS with async-to-LDS copies) x 512
// output columns. Wave: 4 N-tiles, sharing each A fragment across 4 WMMAs.
// Basis kept in registers via complex-rotation recurrence, re-seeded with
// exact integer-phase sincos every CHUNK steps.
// ---------------------------------------------------------------------------
__global__ void __launch_bounds__(256)
irfft_dft_wmma_kernel(const float* __restrict__ A, float* __restrict__ out)
{
    extern __shared__ float sA[];                 // 16 x KPAD floats (131 KB)

    const int tid    = threadIdx.x;
    const int bx     = blockIdx.x;                // 0..127
    const int mstrip = bx >> 2;                   // 0..31  (M strip of 16 rows)
    const int ngrp   = bx & 3;                    // 0..3   (512-column group)

    // ---- stage the contiguous 16 x KPAD A strip into LDS (ASYNCcnt) ----
    {
        const float* gbase = A + (size_t)(mstrip * 16) * KPAD;
        const int nvec = (16 * KPAD) / 4;         // 8208 x b128
        for (int i = tid; i < nvec; i += 256) {
            const int elem = i * 4;
            const uint32_t ldsAddr = (uint32_t)(uintptr_t)(&sA[elem]);
            const float*   gaddr   = gbase + elem;
            asm volatile("global_load_async_to_lds_b128 %0, %1, off"
                         :: "v"(ldsAddr), "v"(gaddr)
                         : "memory");
        }
        asm volatile("s_wait_asynccnt 0" ::: "memory");
    }
    __syncthreads();

    const int lane = tid & 31;
    const int wave = tid >> 5;
    const int half = lane >> 4;                   // 0: K,K+1   1: K+2,K+3
    const int l15  = lane & 15;

    // 4 output columns (time indices) per lane, one per N-tile
    int col[4];
    #pragma unroll
    for (int j = 0; j < 4; ++j)
        col[j] = (ngrp * 32 + wave * 4 + j) * 16 + l15;

    const float TWO_PI_OVER_L = 6.28318530717958647692f / (float)LKER;

    // Constant per-step rotation e^{-i*2*(2*pi*t/L)} for each column
    float rr[4], ri[4];
    #pragma unroll
    for (int j = 0; j < 4; ++j) {
        const int   p2  = (2 * col[j]) & (LKER - 1);
        const float th2 = (float)p2 * TWO_PI_OVER_L;
        float sn2, cs2;
        __sincosf(th2, &sn2, &cs2);
        rr[j] = cs2;
        ri[j] = -sn2;
    }

    const float* aptr = sA + l15 * KPAD + 2 * half;   // LDS, ds_load_b64 per step

    v8f acc0 = {}, acc1 = {}, acc2 = {}, acc3 = {};

    for (int step = 0; step < NSTEP; step += CHUNK) {
        // exact re-seed of the basis phase at l0 = 2*step + half
        const int l0 = 2 * step + half;
        float br[4], bi[4];
        #pragma unroll
        for (int j = 0; j < 4; ++j) {
            const int   p  = (l0 * col[j]) & (LKER - 1);
            const float th = (float)p * TWO_PI_OVER_L;
            float sn, cs;
            __sincosf(th, &sn, &cs);
            br[j] = cs;
            bi[j] = -sn;
        }

        const int send = (step + CHUNK < NSTEP) ? (step + CHUNK) : NSTEP;
        #pragma unroll 2
        for (int s = step; s < send; ++s) {
            const v2f a = *(const v2f*)(aptr + 4 * s);

            v2f b0; b0.x = br[0]; b0.y = bi[0];
            v2f b1; b1.x = br[1]; b1.y = bi[1];
            v2f b2; b2.x = br[2]; b2.y = bi[2];
            v2f b3; b3.x = br[3]; b3.y = bi[3];

            acc0 = __builtin_amdgcn_wmma_f32_16x16x4_f32(false, a, false, b0,
                                                         (short)0, acc0, false, false);
            acc1 = __builtin_amdgcn_wmma_f32_16x16x4_f32(false, a, false, b1,
                                                         (short)0, acc1, false, false);
            acc2 = __builtin_amdgcn_wmma_f32_16x16x4_f32(false, a, false, b2,
                                                         (short)0, acc2, false, false);
            acc3 = __builtin_amdgcn_wmma_f32_16x16x4_f32(false, a, false, b3,
                                                         (short)0, acc3, false, false);

            // advance basis: b *= e^{-i * 2 * (2*pi*t/L)}
            #pragma unroll
            for (int j = 0; j < 4; ++j) {
                const float nr = br[j] * rr[j] - bi[j] * ri[j];
                const float ni = br[j] * ri[j] + bi[j] * rr[j];
                br[j] = nr;
                bi[j] = ni;
            }
        }
    }

    // D layout: VGPR jj -> M = mstrip*16 + jj + 8*half, N = col[j]
    float* obase = out + (size_t)(mstrip * 16 + half * 8) * LKER;
    #pragma unroll
    for (int jj = 0; jj < 8; ++jj) {
        float* orow = obase + (size_t)jj * LKER;
        orow[col[0]] = acc0[jj];
        orow[col[1]] = acc1[jj];
        orow[col[2]] = acc2[jj];
        orow[col[3]] = acc3[jj];
    }
}

// ---------------------------------------------------------------------------
extern "C" void kernel_launch(void* const* d_in, const int* in_sizes, int n_in,
                              void* d_out, int out_size, void* d_ws, size_t ws_size,
                              hipStream_t stream) {
    const float* w_re   = (const float*)d_in[0];
    const float* w_im   = (const float*)d_in[1];
    const float* B_re   = (const float*)d_in[2];
    const float* B_im   = (const float*)d_in[3];
    const float* P_re   = (const float*)d_in[4];
    const float* P_im   = (const float*)d_in[5];
    const float* Q_re   = (const float*)d_in[6];
    const float* Q_im   = (const float*)d_in[7];
    const float* C_re   = (const float*)d_in[8];
    const float* C_im   = (const float*)d_in[9];
    const float* log_dt = (const float*)d_in[10];
    // d_in[11] = L (compile-time constant 2048 here)

    float* Amat = (float*)d_ws;                   // (CHN*HH) x KPAD  ~= 4.2 MB
    float* outp = (float*)d_out;                  // (CHN, HH, LKER) f32

    dim3 g1((LL2 + 255) / 256, HH);
    cauchy_woodbury_kernel<<<g1, dim3(256), 0, stream>>>(
        w_re, w_im, B_re, B_im, P_re, P_im, Q_re, Q_im, C_re, C_im, log_dt, Amat);

    // 32 M-strips x 4 column groups; 8 waves x 4 N-tiles per block
    const size_t ldsBytes = (size_t)16 * KPAD * sizeof(float);   // 131,328 B
    irfft_dft_wmma_kernel<<<dim3(128), dim3(256), ldsBytes, stream>>>(Amat, outp);
}